// WordCountingModule_28123445854739
// MI455X (gfx1250) — compile-verified
//
#include <hip/hip_runtime.h>
#include <stdint.h>

#define VOCAB 32000
#define WAVES_PER_BLOCK 8
#define THREADS (WAVES_PER_BLOCK * 32)
#define CHUNK_FLOATS 128                    // 32 lanes * 4 floats (b128 per lane)
#define NCHUNK (VOCAB / CHUNK_FLOATS)       // 250
#define DEPTH 4                             // async pipeline depth (power of 2)

// ---------------- CDNA5 async global->LDS helpers (inline asm, CDNA5 ISA) ----

__device__ __forceinline__ void async_load_b128_nt(uint32_t lds_off, const float* gp) {
    // GV mode: 64-bit address in a VGPR pair, LDS destination offset in a VGPR.
    // Tracked with ASYNCcnt; per-lane 16B transfer, lanes contiguous -> 512B/wave.
    // th:TH_LOAD_NT: stream is read exactly once -> non-temporal, don't churn L2/WGP$.
    asm volatile("global_load_async_to_lds_b128 %0, %1, off th:TH_LOAD_NT"
                 :: "v"(lds_off), "v"((uint64_t)(uintptr_t)gp)
                 : "memory");
}

template <int N>
__device__ __forceinline__ void wait_async() {
    asm volatile("s_wait_asynccnt %0" :: "i"(N) : "memory");
}

// ---------------- main kernel first so the disasm snippet shows it -------------------

__global__ __launch_bounds__(THREADS)
void wc_argmax_hist(const float* __restrict__ utt, float* __restrict__ out, int nrows) {
    __shared__ float lds[WAVES_PER_BLOCK * DEPTH * CHUNK_FLOATS];  // 16 KB

    const int lane = threadIdx.x & 31;
    const int wave = threadIdx.x >> 5;
    const int row  = blockIdx.x * WAVES_PER_BLOCK + wave;
    if (row >= nrows) return;  // uniform per wave; no barriers used below

    const float* gbase = utt + (size_t)row * VOCAB + lane * 4;
    float*       mybuf = &lds[(wave * DEPTH) * CHUNK_FLOATS];

    // Per-lane LDS byte offsets for each pipeline slot (low 32 bits of the
    // generic pointer == LDS offset; HW adds the wave's LDS_BASE).
    uint32_t lds_off[DEPTH];
#pragma unroll
    for (int s = 0; s < DEPTH; ++s)
        lds_off[s] = (uint32_t)(uintptr_t)(&mybuf[s * CHUNK_FLOATS + lane * 4]);

    float best = -__builtin_inff();
    int   bidx = 0;

    // Prefill DEPTH-1 chunks.
    async_load_b128_nt(lds_off[0], gbase + (size_t)0 * CHUNK_FLOATS);
    async_load_b128_nt(lds_off[1], gbase + (size_t)1 * CHUNK_FLOATS);
    async_load_b128_nt(lds_off[2], gbase + (size_t)2 * CHUNK_FLOATS);

    // Steady state: issue chunk i+3, wait ASYNCcnt<=3 (async loads complete in
    // order -> oldest outstanding chunk i is resident), consume chunk i from LDS.
    for (int i = 0; i < NCHUNK - (DEPTH - 1); ++i) {
        const int nxt = i + (DEPTH - 1);
        async_load_b128_nt(lds_off[nxt & (DEPTH - 1)],
                           gbase + (size_t)nxt * CHUNK_FLOATS);
        wait_async<DEPTH - 1>();

        const float4 v = *reinterpret_cast<const float4*>(
            &mybuf[(i & (DEPTH - 1)) * CHUNK_FLOATS + lane * 4]);
        const int base = i * CHUNK_FLOATS + lane * 4;
        if (v.x > best) { best = v.x; bidx = base;     }
        if (v.y > best) { best = v.y; bidx = base + 1; }
        if (v.z > best) { best = v.z; bidx = base + 2; }
        if (v.w > best) { best = v.w; bidx = base + 3; }
    }

    // Tail: drain pipeline with descending wait immediates.
#pragma unroll
    for (int t = DEPTH - 2; t >= 0; --t) {
        const int i = NCHUNK - 1 - t;
        if      (t == 2) wait_async<2>();
        else if (t == 1) wait_async<1>();
        else             wait_async<0>();

        const float4 v = *reinterpret_cast<const float4*>(
            &mybuf[(i & (DEPTH - 1)) * CHUNK_FLOATS + lane * 4]);
        const int base = i * CHUNK_FLOATS + lane * 4;
        if (v.x > best) { best = v.x; bidx = base;     }
        if (v.y > best) { best = v.y; bidx = base + 1; }
        if (v.z > best) { best = v.z; bidx = base + 2; }
        if (v.w > best) { best = v.w; bidx = base + 3; }
    }

    // Wave32 (val,idx) reduction; lower index wins ties (matches jnp.argmax).
#pragma unroll
    for (int off = 16; off > 0; off >>= 1) {
        float ov = __shfl_xor(best, off, 32);
        int   oi = __shfl_xor(bidx, off, 32);
        if (ov > best || (ov == best && oi < bidx)) { best = ov; bidx = oi; }
    }

    if (lane == 0) atomicAdd(&out[bidx], 1.0f);
}

// ---------------- output init (runs first on the stream) ----------------------------

__global__ void wc_init_out(const float* __restrict__ wc, float* __restrict__ out) {
    int i = blockIdx.x * blockDim.x + threadIdx.x;
    if (i < VOCAB) out[i] = wc[i];
}

// ---------------- launch -------------------------------------------------------------

extern "C" void kernel_launch(void* const* d_in, const int* in_sizes, int n_in,
                              void* d_out, int out_size, void* d_ws, size_t ws_size,
                              hipStream_t stream) {
    const float* utt = (const float*)d_in[0];   // [2048, 4, 32000] fp32
    const float* wc  = (const float*)d_in[1];   // [32000] fp32
    float*       out = (float*)d_out;           // [32000] fp32

    const int nrows = in_sizes[0] / VOCAB;      // 8192

    wc_init_out<<<(VOCAB + 255) / 256, 256, 0, stream>>>(wc, out);
    wc_argmax_hist<<<(nrows + WAVES_PER_BLOCK - 1) / WAVES_PER_BLOCK,
                     THREADS, 0, stream>>>(utt, out, nrows);
}